// GAT_43550968382059
// MI455X (gfx1250) — compile-verified
//
#include <hip/hip_runtime.h>
#include <math.h>

#define NN 50000
#define EE 800000
#define ET (EE + NN)          // edges + self loops
#define IN_DIM 128
#define HID 32
#define HEADS 4
#define HC 128                // HEADS*HID
#define OUT_DIM 40
#define SLOPE 0.2f

typedef float v2f __attribute__((ext_vector_type(2)));
typedef float v8f __attribute__((ext_vector_type(8)));

// ---------------- order-preserving float<->uint for atomic max ----------------
__device__ __forceinline__ unsigned ford(float f) {
    unsigned u = __float_as_uint(f);
    return (u & 0x80000000u) ? ~u : (u | 0x80000000u);
}
__device__ __forceinline__ float funord(unsigned k) {
    unsigned u = (k & 0x80000000u) ? (k & 0x7FFFFFFFu) : ~k;
    return __uint_as_float(u);
}
#define FORD_NEG_INF 0x007FFFFFu   // ford(-inf)

__device__ __forceinline__ void edge_sd(int u, const int* __restrict__ ei,
                                        int& src, int& dst) {
    if (u < EE) { src = ei[u]; dst = ei[EE + u]; }
    else        { src = u - EE; dst = u - EE; }     // self loop
}

__device__ __forceinline__ float leaky(float e) {
    return e > 0.f ? e : SLOPE * e;
}

__device__ __forceinline__ v8f wmma4(v2f a, v2f b, v8f c) {
    return __builtin_amdgcn_wmma_f32_16x16x4_f32(
               false, a, false, b, (short)0, c, false, false);
}

// =====================================================================
// WMMA f32 GEMM, fast path: C[M x Nout] = A[M x K] @ B[K x Nout]
// one wave32 per 16x32 output tile (two accumulators, shared A load).
// Requires: M % 16 == 0, K % 4 == 0, Nout % 32 == 0. Branch-free loop.
// =====================================================================
__global__ __launch_bounds__(32) void wmma_gemm_f32_2t(
    const float* __restrict__ A, const float* __restrict__ B,
    float* __restrict__ C, int K, int Nout)
{
    const int mt   = blockIdx.x;
    const int nt   = blockIdx.y;       // 32-col tile
    const int lane = threadIdx.x;
    const int hi   = lane >> 4;        // 0|1
    const int lo   = lane & 15;

    const float* Ap = A + (size_t)(mt * 16 + lo) * K + 2 * hi;        // K=+0,+1 / +2,+3
    const float* Bp = B + (size_t)(2 * hi) * Nout + nt * 32 + lo;

    v8f acc0 = {}, acc1 = {};
    for (int k = 0; k < K; k += 4) {
        float2 av = *(const float2*)(Ap + k);
        v2f a; a.x = av.x; a.y = av.y;
        const float* bp = Bp + (size_t)k * Nout;
        v2f b0; b0.x = bp[0];  b0.y = bp[Nout];
        v2f b1; b1.x = bp[16]; b1.y = bp[Nout + 16];
        acc0 = wmma4(a, b0, acc0);
        acc1 = wmma4(a, b1, acc1);
    }
    const int col = nt * 32 + lo;
    float* Cp = C + (size_t)(mt * 16 + 8 * hi) * Nout + col;
    #pragma unroll
    for (int r = 0; r < 8; ++r) {                 // lanes16-31 hold M=r+8
        Cp[(size_t)r * Nout + 0]  = acc0[r];
        Cp[(size_t)r * Nout + 16] = acc1[r];
    }
}

// =====================================================================
// WMMA f32 GEMM, tail path (Nout not multiple of 16): one wave per 16x16
// tile; out-of-range lanes clamp their load address (a B column only
// feeds the same D column, owned by that lane) and skip the store.
// =====================================================================
__global__ __launch_bounds__(32) void wmma_gemm_f32_tail(
    const float* __restrict__ A, const float* __restrict__ B,
    float* __restrict__ C, const float* __restrict__ bias,
    int K, int Nout)
{
    const int mt   = blockIdx.x;
    const int nt   = blockIdx.y;
    const int lane = threadIdx.x;
    const int hi   = lane >> 4;
    const int lo   = lane & 15;
    const int col  = nt * 16 + lo;
    const int ccol = col < Nout ? col : Nout - 1;    // clamped, always in-bounds

    const float* Ap = A + (size_t)(mt * 16 + lo) * K + 2 * hi;
    const float* Bp = B + (size_t)(2 * hi) * Nout + ccol;

    v8f acc = {};
    for (int k = 0; k < K; k += 4) {
        float2 av = *(const float2*)(Ap + k);
        v2f a; a.x = av.x; a.y = av.y;
        const float* bp = Bp + (size_t)k * Nout;
        v2f b; b.x = bp[0]; b.y = bp[Nout];
        acc = wmma4(a, b, acc);
    }
    if (col < Nout) {
        const float bb = bias ? bias[col] : 0.f;
        float* Cp = C + (size_t)(mt * 16 + 8 * hi) * Nout + col;
        #pragma unroll
        for (int r = 0; r < 8; ++r)
            Cp[(size_t)r * Nout] = acc[r] + bb;
    }
}

// ---------------- per-(node,head) attention logits ----------------
__global__ void attn_coef(const float* __restrict__ h,
                          const float* __restrict__ a_src,
                          const float* __restrict__ a_dst,
                          float* __restrict__ al_s,
                          float* __restrict__ al_d)
{
    int idx = blockIdx.x * blockDim.x + threadIdx.x;   // n*HEADS + head
    if (idx >= NN * HEADS) return;
    int n = idx / HEADS, hd = idx % HEADS;
    const float* hp = h + (size_t)n * HC + hd * HID;
    const float* as = a_src + hd * HID;
    const float* ad = a_dst + hd * HID;
    float s = 0.f, d = 0.f;
    #pragma unroll
    for (int c = 0; c < HID; ++c) { float v = hp[c]; s = fmaf(v, as[c], s); d = fmaf(v, ad[c], d); }
    al_s[idx] = s; al_d[idx] = d;
}

// ---------------- init: m = -inf (ordered), den = 0, acc = 0 ----------------
__global__ void init_layer(unsigned* __restrict__ m, float* __restrict__ den,
                           float* __restrict__ acc)
{
    int i = blockIdx.x * blockDim.x + threadIdx.x;
    if (i < NN * HEADS) { m[i] = FORD_NEG_INF; den[i] = 0.f; }
    if (i < NN * HC)    acc[i] = 0.f;
}

// ---------------- edge pass 1: segment max ----------------
__global__ void edge_max(const int* __restrict__ ei,
                         const float* __restrict__ al_s,
                         const float* __restrict__ al_d,
                         unsigned* __restrict__ m)
{
    int idx = blockIdx.x * blockDim.x + threadIdx.x;
    if (idx >= ET * HEADS) return;
    int u = idx >> 2, hd = idx & 3;
    int src, dst; edge_sd(u, ei, src, dst);
    float e = leaky(al_s[src * HEADS + hd] + al_d[dst * HEADS + hd]);
    atomicMax(&m[dst * HEADS + hd], ford(e));
}

// ---------------- edge pass 2: segment sum of exp ----------------
__global__ void edge_sum(const int* __restrict__ ei,
                         const float* __restrict__ al_s,
                         const float* __restrict__ al_d,
                         const unsigned* __restrict__ m,
                         float* __restrict__ den)
{
    int idx = blockIdx.x * blockDim.x + threadIdx.x;
    if (idx >= ET * HEADS) return;
    int u = idx >> 2, hd = idx & 3;
    int src, dst; edge_sd(u, ei, src, dst);
    float e  = leaky(al_s[src * HEADS + hd] + al_d[dst * HEADS + hd]);
    float mx = funord(m[dst * HEADS + hd]);
    atomicAdd(&den[dst * HEADS + hd], __expf(e - mx));
}

// ---------------- edge pass 3: weighted gather/scatter, one wave per edge ----------------
__global__ __launch_bounds__(256) void edge_aggregate(
    const int* __restrict__ ei, const float* __restrict__ h,
    const float* __restrict__ al_s, const float* __restrict__ al_d,
    const unsigned* __restrict__ m, const float* __restrict__ den,
    float* __restrict__ acc)
{
    int wave = blockIdx.x * (blockDim.x >> 5) + (threadIdx.x >> 5);
    int lane = threadIdx.x & 31;
    if (wave >= ET) return;
    int src, dst; edge_sd(wave, ei, src, dst);
    int hd = lane >> 3;                         // 4 channels/lane -> head = lane/8
    float e  = leaky(al_s[src * HEADS + hd] + al_d[dst * HEADS + hd]);
    float mx = funord(m[dst * HEADS + hd]);
    float alpha = __expf(e - mx) / (den[dst * HEADS + hd] + 1e-16f);

    float4 v = *((const float4*)(h + (size_t)src * HC) + lane);
    float* ap = acc + (size_t)dst * HC + lane * 4;
    atomicAdd(ap + 0, v.x * alpha);
    atomicAdd(ap + 1, v.y * alpha);
    atomicAdd(ap + 2, v.z * alpha);
    atomicAdd(ap + 3, v.w * alpha);
}

// ---------------- bias + relu epilogue (in place) ----------------
__global__ void bias_relu(float* __restrict__ acc, const float* __restrict__ b)
{
    int i = blockIdx.x * blockDim.x + threadIdx.x;
    if (i >= NN * HC) return;
    float v = acc[i] + b[i & (HC - 1)];
    acc[i] = v > 0.f ? v : 0.f;
}

// =======================================================================
extern "C" void kernel_launch(void* const* d_in, const int* in_sizes, int n_in,
                              void* d_out, int out_size, void* d_ws, size_t ws_size,
                              hipStream_t stream)
{
    const float* x   = (const float*)d_in[0];
    const int*   ei  = (const int*)d_in[1];
    const float* W[3]    = { (const float*)d_in[2],  (const float*)d_in[6],  (const float*)d_in[10] };
    const float* asv[3]  = { (const float*)d_in[3],  (const float*)d_in[7],  (const float*)d_in[11] };
    const float* adv[3]  = { (const float*)d_in[4],  (const float*)d_in[8],  (const float*)d_in[12] };
    const float* bv[3]   = { (const float*)d_in[5],  (const float*)d_in[9],  (const float*)d_in[13] };
    const float* Wl  = (const float*)d_in[14];
    const float* bl  = (const float*)d_in[15];

    // workspace layout
    char* ws = (char*)d_ws;
    float*    hbuf = (float*)ws;                       ws += (size_t)NN * HC * sizeof(float);
    float*    bufA = (float*)ws;                       ws += (size_t)NN * HC * sizeof(float);
    float*    bufB = (float*)ws;                       ws += (size_t)NN * HC * sizeof(float);
    float*    al_s = (float*)ws;                       ws += (size_t)NN * HEADS * sizeof(float);
    float*    al_d = (float*)ws;                       ws += (size_t)NN * HEADS * sizeof(float);
    unsigned* mbuf = (unsigned*)ws;                    ws += (size_t)NN * HEADS * sizeof(unsigned);
    float*    den  = (float*)ws;

    const int TPB = 256;
    const int MT  = NN / 16;                 // 3125 row tiles (exact)
    const int gNHC   = (NN * HC + TPB - 1) / TPB;
    const int gNH    = (NN * HEADS + TPB - 1) / TPB;
    const int gEH    = (ET * HEADS + TPB - 1) / TPB;
    const int gWaves = (ET + (TPB / 32) - 1) / (TPB / 32);

    auto gat_layer = [&](const float* xin, int layer, float* acc) {
        hipLaunchKernelGGL(init_layer, dim3(gNHC), dim3(TPB), 0, stream, mbuf, den, acc);
        hipLaunchKernelGGL(wmma_gemm_f32_2t, dim3(MT, HC / 32), dim3(32), 0, stream,
                           xin, W[layer], hbuf, IN_DIM, HC);
        hipLaunchKernelGGL(attn_coef, dim3(gNH), dim3(TPB), 0, stream,
                           hbuf, asv[layer], adv[layer], al_s, al_d);
        hipLaunchKernelGGL(edge_max, dim3(gEH), dim3(TPB), 0, stream, ei, al_s, al_d, mbuf);
        hipLaunchKernelGGL(edge_sum, dim3(gEH), dim3(TPB), 0, stream, ei, al_s, al_d, mbuf, den);
        hipLaunchKernelGGL(edge_aggregate, dim3(gWaves), dim3(TPB), 0, stream,
                           ei, hbuf, al_s, al_d, mbuf, den, acc);
        hipLaunchKernelGGL(bias_relu, dim3(gNHC), dim3(TPB), 0, stream, acc, bv[layer]);
    };

    gat_layer(x,    0, bufA);    // x1
    gat_layer(bufA, 1, bufB);    // x2
    gat_layer(bufB, 2, bufA);    // x3

    // head: out = x3 @ Wl + bl   (40 cols -> 3 guarded col tiles)
    hipLaunchKernelGGL(wmma_gemm_f32_tail, dim3(MT, (OUT_DIM + 15) / 16), dim3(32), 0, stream,
                       bufA, Wl, (float*)d_out, bl, HC, OUT_DIM);
}